// GCN4_1348619731442
// MI455X (gfx1250) — compile-verified
//
#include <hip/hip_runtime.h>

typedef __attribute__((ext_vector_type(2))) float v2f;
typedef __attribute__((ext_vector_type(8))) float v8f;

#define NNODES 50000
#define NEDGES 800000
#define F_IN   128
#define H1     256
#define H2     128
#define H3     64
#define NCLS   40
#define MT     (NNODES / 16)   // 3125 row tiles, exact
#define KCHUNK 128
#define LDSTR  (KCHUNK + 4)    // stride 132 dwords: 132 % 64 == 4 -> conflict-free b64 reads

// ---------------- degree / CSR precompute ----------------

__global__ void node_init_kernel(float* __restrict__ deg, int* __restrict__ counts) {
    int i = blockIdx.x * blockDim.x + threadIdx.x;
    if (i < NNODES) { deg[i] = 1.0f; counts[i] = 0; }   // deg starts at 1 (self-loop)
}

__global__ void edge_count_kernel(const int* __restrict__ dst,
                                  float* __restrict__ deg, int* __restrict__ counts) {
    int e = blockIdx.x * blockDim.x + threadIdx.x;
    if (e < NEDGES) {
        int d = dst[e];
        atomicAdd(&deg[d], 1.0f);
        atomicAdd(&counts[d], 1);
    }
}

__global__ void deg_to_dinv_kernel(float* __restrict__ deg) {
    int i = blockIdx.x * blockDim.x + threadIdx.x;
    if (i < NNODES) deg[i] = rsqrtf(deg[i]);            // deg >= 1 always
}

// single-block exclusive scan of counts[NNODES] -> offsets[NNODES+1]
__global__ void scan_kernel(const int* __restrict__ counts, int* __restrict__ offsets) {
    __shared__ int sm[1024];
    __shared__ int carry;
    int tid = threadIdx.x;
    if (tid == 0) carry = 0;
    __syncthreads();
    for (int base = 0; base < NNODES; base += 1024) {
        int i = base + tid;
        int v = (i < NNODES) ? counts[i] : 0;
        sm[tid] = v;
        __syncthreads();
        for (int off = 1; off < 1024; off <<= 1) {
            int t = (tid >= off) ? sm[tid - off] : 0;
            __syncthreads();
            sm[tid] += t;
            __syncthreads();
        }
        int incl = sm[tid];
        int c = carry;
        if (i < NNODES) offsets[i] = c + incl - v;      // exclusive prefix
        __syncthreads();
        if (tid == 1023) carry = c + incl;
        __syncthreads();
    }
    if (tid == 0) offsets[NNODES] = carry;
}

__global__ void cursor_init_kernel(const int* __restrict__ offsets, int* __restrict__ cursor) {
    int i = blockIdx.x * blockDim.x + threadIdx.x;
    if (i < NNODES) cursor[i] = offsets[i];
}

// bin edges by destination; store source id and precomputed norm weight contiguously
__global__ void place_kernel(const int* __restrict__ src, const int* __restrict__ dst,
                             const float* __restrict__ dinv, int* __restrict__ cursor,
                             int* __restrict__ esrc, float* __restrict__ enorm) {
    int e = blockIdx.x * blockDim.x + threadIdx.x;
    if (e < NEDGES) {
        int s = src[e], d = dst[e];
        int pos = atomicAdd(&cursor[d], 1);
        esrc[pos]  = s;
        enorm[pos] = dinv[s] * dinv[d];
    }
}

// ---------------- fp32 WMMA GEMM: out[M,Fout] = act(A[M,K]) @ W[K,Fout] ----------------
// Block = (32,8): 8 waves. Each wave computes a 16x64 strip (4 WMMA tiles sharing the A
// fragment). The block stages a 64-column slab of W in LDS (column-major, stride 132)
// in K-chunks of 128; B fragments come from LDS as bank-conflict-free ds_load_b64.
__global__ void gemm_wmma_f32_kernel(const float* __restrict__ A, const float* __restrict__ W,
                                     float* __restrict__ out, int K, int Fout, int reluIn) {
    __shared__ float wlds[64 * LDSTR];       // 33,792 B

    int colBase = blockIdx.y * 64;
    int tid     = threadIdx.y * 32 + threadIdx.x;
    int tileM   = blockIdx.x * 8 + threadIdx.y;
    bool active = tileM < MT;

    int lane = threadIdx.x;
    int kh   = lane >> 4;                    // K-half selector (0/1)
    int l    = lane & 15;
    int row  = (active ? tileM : 0) * 16 + l;
    const float* arow = A + (size_t)row * K + kh * 2;

    v8f acc[4] = {v8f{}, v8f{}, v8f{}, v8f{}};

    for (int k0 = 0; k0 < K; k0 += KCHUNK) {
        int klen = K - k0; if (klen > KCHUNK) klen = KCHUNK;
        __syncthreads();                     // protect previous chunk's reads
        // cooperative load: coalesced on global, W^T layout in LDS
        for (int idx = tid; idx < 64 * KCHUNK; idx += 256) {
            int r = idx >> 6;                // K row within chunk
            int c = idx & 63;                // local column
            int col = colBase + c;
            float v = 0.0f;
            if (r < klen && col < Fout) v = W[(size_t)(k0 + r) * Fout + col];
            wlds[c * LDSTR + r] = v;
        }
        __syncthreads();

        if (active) {
            for (int kk = 0; kk < klen; kk += 4) {
                float2 av = *(const float2*)(arow + k0 + kk);   // rows of A, K pair per half
                v2f a; a.x = av.x; a.y = av.y;
                if (reluIn) { a.x = fmaxf(a.x, 0.0f); a.y = fmaxf(a.y, 0.0f); }

                int kb = kk + kh * 2;        // lanes 0-15: K=kk..kk+1, lanes 16-31: kk+2..kk+3
                #pragma unroll
                for (int t = 0; t < 4; ++t) {
                    float2 bv = *(const float2*)(wlds + (t * 16 + l) * LDSTR + kb); // ds_load_b64
                    v2f b; b.x = bv.x; b.y = bv.y;
                    acc[t] = __builtin_amdgcn_wmma_f32_16x16x4_f32(
                        false, a, false, b, (short)0, acc[t], false, false);
                }
            }
        }
    }

    if (active) {
        int rbase = tileM * 16 + kh * 8;     // C/D layout: vgpr r -> row r (+8 for high lanes)
        #pragma unroll
        for (int t = 0; t < 4; ++t) {
            int col = colBase + t * 16 + l;
            if (col < Fout) {
                #pragma unroll
                for (int r = 0; r < 8; ++r)
                    out[(size_t)(rbase + r) * Fout + col] = acc[t][r];
            }
        }
    }
}

// ---------------- gather-style aggregation (no fp atomics) ----------------
// One wave per destination node; F features striped across 32 lanes, accumulated
// in registers. Self-loop + bias fused. out[d,:] = sum_e w_e * t[src_e,:] + w_self*t[d,:] + b
template <int F>
__global__ void agg_gather_kernel(const float* __restrict__ t, const int* __restrict__ offs,
                                  const int* __restrict__ esrc, const float* __restrict__ enorm,
                                  const float* __restrict__ dinv, const float* __restrict__ bias,
                                  float* __restrict__ out) {
    int node = blockIdx.x * 8 + threadIdx.y;
    if (node >= NNODES) return;
    int lane = threadIdx.x;
    constexpr int NR = (F + 31) / 32;

    float di = dinv[node];
    float wself = di * di;
    float acc[NR];
    #pragma unroll
    for (int j = 0; j < NR; ++j) {
        int f = lane + j * 32;
        acc[j] = (f < F) ? wself * t[(size_t)node * F + f] : 0.0f;
    }

    int p0 = offs[node], p1 = offs[node + 1];
    for (int p = p0; p < p1; ++p) {
        int s   = esrc[p];
        float w = enorm[p];
        const float* rowp = t + (size_t)s * F;
        if (p + 1 < p1)
            __builtin_prefetch(t + (size_t)esrc[p + 1] * F + lane, 0, 3);  // global_prefetch_b8
        #pragma unroll
        for (int j = 0; j < NR; ++j) {
            int f = lane + j * 32;
            if (f < F) acc[j] = fmaf(w, rowp[f], acc[j]);
        }
    }

    #pragma unroll
    for (int j = 0; j < NR; ++j) {
        int f = lane + j * 32;
        if (f < F) out[(size_t)node * F + f] = acc[j] + bias[f];
    }
}

// ---------------- launcher ----------------

static inline int cdiv(long long a, long long b) { return (int)((a + b - 1) / b); }

extern "C" void kernel_launch(void* const* d_in, const int* in_sizes, int n_in,
                              void* d_out, int out_size, void* d_ws, size_t ws_size,
                              hipStream_t stream) {
    (void)in_sizes; (void)n_in; (void)out_size; (void)ws_size;

    const float* x  = (const float*)d_in[0];
    const int*   ei = (const int*)d_in[1];
    const float* W1 = (const float*)d_in[2];
    const float* b1 = (const float*)d_in[3];
    const float* W2 = (const float*)d_in[4];
    const float* b2 = (const float*)d_in[5];
    const float* W3 = (const float*)d_in[6];
    const float* b3 = (const float*)d_in[7];
    const float* W4 = (const float*)d_in[8];
    const float* b4 = (const float*)d_in[9];
    float* outp = (float*)d_out;

    const int* src = ei;            // edge_index[0]
    const int* dst = ei + NEDGES;   // edge_index[1]

    // workspace layout: dinv[N] f32 | offsets[N+1] i32 | counts/cursor[N] i32 |
    //                   esrc[E] i32 | enorm[E] f32 | bufT[N*256] f32 | bufA[N*256] f32
    float* dinv    = (float*)d_ws;
    int*   offsets = (int*)(dinv + NNODES);
    int*   counts  = offsets + (NNODES + 1);     // reused as cursor after scan
    int*   esrc    = counts + NNODES;
    float* enorm   = (float*)(esrc + NEDGES);
    float* bufT    = enorm + NEDGES;
    float* bufA    = bufT + (size_t)NNODES * H1;

    const dim3 gblk(32, 8);

    // --- degree + CSR-by-destination build ---
    node_init_kernel<<<cdiv(NNODES, 256), 256, 0, stream>>>(dinv, counts);
    edge_count_kernel<<<cdiv(NEDGES, 256), 256, 0, stream>>>(dst, dinv, counts);
    deg_to_dinv_kernel<<<cdiv(NNODES, 256), 256, 0, stream>>>(dinv);
    scan_kernel<<<1, 1024, 0, stream>>>(counts, offsets);
    cursor_init_kernel<<<cdiv(NNODES, 256), 256, 0, stream>>>(offsets, counts);
    place_kernel<<<cdiv(NEDGES, 256), 256, 0, stream>>>(src, dst, dinv, counts, esrc, enorm);

    const int aggBlocks = cdiv(NNODES, 8);

    // --- layer 1: x[50000,128] @ W1[128,256] -> gather-agg (+b1); ReLU fused into next GEMM
    gemm_wmma_f32_kernel<<<dim3(cdiv(MT, 8), H1 / 64), gblk, 0, stream>>>(x, W1, bufT, F_IN, H1, 0);
    agg_gather_kernel<H1><<<aggBlocks, gblk, 0, stream>>>(bufT, offsets, esrc, enorm, dinv, b1, bufA);

    // --- layer 2: relu(h1)[.,256] @ W2[256,128] ---
    gemm_wmma_f32_kernel<<<dim3(cdiv(MT, 8), H2 / 64), gblk, 0, stream>>>(bufA, W2, bufT, H1, H2, 1);
    agg_gather_kernel<H2><<<aggBlocks, gblk, 0, stream>>>(bufT, offsets, esrc, enorm, dinv, b2, bufA);

    // --- layer 3: relu(h2)[.,128] @ W3[128,64] ---
    gemm_wmma_f32_kernel<<<dim3(cdiv(MT, 8), 1), gblk, 0, stream>>>(bufA, W3, bufT, H2, H3, 1);
    agg_gather_kernel<H3><<<aggBlocks, gblk, 0, stream>>>(bufT, offsets, esrc, enorm, dinv, b3, bufA);

    // --- layer 4: relu(h3)[.,64] @ W4[64,40] -> gather-agg (+b4, no ReLU) -> d_out ---
    gemm_wmma_f32_kernel<<<dim3(cdiv(MT, 8), 1), gblk, 0, stream>>>(bufA, W4, bufT, H3, NCLS, 1);
    agg_gather_kernel<NCLS><<<aggBlocks, gblk, 0, stream>>>(bufT, offsets, esrc, enorm, dinv, b4, outp);
}